// SioConvLayer_88527865905508
// MI455X (gfx1250) — compile-verified
//
#include <hip/hip_runtime.h>
#include <hip/hip_bf16.h>
#include <math.h>

// ---------------------------------------------------------------------------
// SioConv forward for MI455X (gfx1250, wave32).
// B=4, L=1024, D=512, H=8, I=64.
//
// Reference's tril-over-(m,h) mask collapses the O(L^2) einsum:
//   h[b,l,h,i] = z[b,h,h,i] + exp(S[b,l,h]) * (sum_{m>h} z[b,m,h,i] + hidden[b,h,i])
// with S = inclusive cumsum_l(ln_da).  Dominant work = 4 f32 GEMMs
// [4096x512]x[512x512], done with V_WMMA_F32_16X16X4_F32.
// ---------------------------------------------------------------------------

#define Bsz 4
#define Lsz 1024
#define Dsz 512
#define Hsz 8
#define Isz 64
#define Msz (Bsz * Lsz)   // 4096
#define GN_EPS 1e-5f

typedef float v2f __attribute__((ext_vector_type(2)));
typedef float v8f __attribute__((ext_vector_type(8)));

__device__ __forceinline__ v8f wmma4(v2f a, v2f b, v8f c) {
  // D = A(16x4,f32) * B(4x16,f32) + C(16x16,f32)
  return __builtin_amdgcn_wmma_f32_16x16x4_f32(
      /*neg_a=*/false, a, /*neg_b=*/false, b,
      /*c_mod=*/(short)0, c, /*reuse_a=*/false, /*reuse_b=*/false);
}

__device__ __forceinline__ float silu_f(float v) {
  return v / (1.0f + expf(-v));
}

// ---------------------------------------------------------------------------
// Kernel 1: Z = (X @ Wz^T + bz) * silu(X @ Wza^T + bza)
// Block = 128 threads (4 waves), each wave a 32x32 output tile.
// Grid = (M/64, D/64).
// ---------------------------------------------------------------------------
__global__ __launch_bounds__(128)
void gemm_gate_z_kernel(const float* __restrict__ X,
                        const float* __restrict__ Wz,  const float* __restrict__ bz,
                        const float* __restrict__ Wza, const float* __restrict__ bza,
                        float* __restrict__ Z) {
  const int lane = threadIdx.x & 31;
  const int wave = threadIdx.x >> 5;
  const int mr = lane & 15;     // A: M row within 16; B: N col within 16
  const int kk = lane >> 4;     // selects K-pair {2kk, 2kk+1}
  const int m_base = blockIdx.x * 64 + (wave & 1) * 32;
  const int n_base = blockIdx.y * 64 + (wave >> 1) * 32;

  v8f a1[2][2], a2[2][2];
  const v8f zero = {0.f, 0.f, 0.f, 0.f, 0.f, 0.f, 0.f, 0.f};
  for (int i = 0; i < 2; ++i)
    for (int j = 0; j < 2; ++j) { a1[i][j] = zero; a2[i][j] = zero; }

  const float* A0  = X   + (size_t)(m_base + mr)       * Dsz + 2 * kk;
  const float* A1  = A0  + 16 * Dsz;
  const float* Bz0 = Wz  + (size_t)(n_base + mr)       * Dsz + 2 * kk;
  const float* Bz1 = Bz0 + 16 * Dsz;
  const float* Ba0 = Wza + (size_t)(n_base + mr)       * Dsz + 2 * kk;
  const float* Ba1 = Ba0 + 16 * Dsz;

#pragma unroll 4
  for (int k0 = 0; k0 < Dsz; k0 += 4) {
    v2f fa0 = *(const v2f*)(A0 + k0);
    v2f fa1 = *(const v2f*)(A1 + k0);
    v2f bz0 = *(const v2f*)(Bz0 + k0);
    v2f bz1 = *(const v2f*)(Bz1 + k0);
    v2f ba0 = *(const v2f*)(Ba0 + k0);
    v2f ba1 = *(const v2f*)(Ba1 + k0);
    a1[0][0] = wmma4(fa0, bz0, a1[0][0]);
    a1[0][1] = wmma4(fa0, bz1, a1[0][1]);
    a1[1][0] = wmma4(fa1, bz0, a1[1][0]);
    a1[1][1] = wmma4(fa1, bz1, a1[1][1]);
    a2[0][0] = wmma4(fa0, ba0, a2[0][0]);
    a2[0][1] = wmma4(fa0, ba1, a2[0][1]);
    a2[1][0] = wmma4(fa1, ba0, a2[1][0]);
    a2[1][1] = wmma4(fa1, ba1, a2[1][1]);
  }

  for (int mt = 0; mt < 2; ++mt)
    for (int nt = 0; nt < 2; ++nt) {
      const int n = n_base + nt * 16 + mr;
      const float biz = bz[n], biza = bza[n];
#pragma unroll
      for (int r = 0; r < 8; ++r) {
        const int m = m_base + mt * 16 + kk * 8 + r;
        const float v1 = a1[mt][nt][r] + biz;
        const float v2 = a2[mt][nt][r] + biza;
        Z[(size_t)m * Dsz + n] = v1 * silu_f(v2);
      }
    }
}

// ---------------------------------------------------------------------------
// Kernel 2 (templated epilogue): OUT = f(A @ W^T + bias)
//   MODE 0: f(v) = silu(v)                  (g_ya)
//   MODE 1: f(v) = v * gate[m*D + n]        (y)
// ---------------------------------------------------------------------------
template <int MODE>
__global__ __launch_bounds__(128)
void gemm_single_kernel(const float* __restrict__ A, const float* __restrict__ W,
                        const float* __restrict__ bias,
                        const float* __restrict__ gate, float* __restrict__ OUT) {
  const int lane = threadIdx.x & 31;
  const int wave = threadIdx.x >> 5;
  const int mr = lane & 15;
  const int kk = lane >> 4;
  const int m_base = blockIdx.x * 64 + (wave & 1) * 32;
  const int n_base = blockIdx.y * 64 + (wave >> 1) * 32;

  v8f acc[2][2];
  const v8f zero = {0.f, 0.f, 0.f, 0.f, 0.f, 0.f, 0.f, 0.f};
  for (int i = 0; i < 2; ++i)
    for (int j = 0; j < 2; ++j) acc[i][j] = zero;

  const float* A0 = A + (size_t)(m_base + mr) * Dsz + 2 * kk;
  const float* A1 = A0 + 16 * Dsz;
  const float* B0 = W + (size_t)(n_base + mr) * Dsz + 2 * kk;
  const float* B1 = B0 + 16 * Dsz;

#pragma unroll 4
  for (int k0 = 0; k0 < Dsz; k0 += 4) {
    v2f fa0 = *(const v2f*)(A0 + k0);
    v2f fa1 = *(const v2f*)(A1 + k0);
    v2f fb0 = *(const v2f*)(B0 + k0);
    v2f fb1 = *(const v2f*)(B1 + k0);
    acc[0][0] = wmma4(fa0, fb0, acc[0][0]);
    acc[0][1] = wmma4(fa0, fb1, acc[0][1]);
    acc[1][0] = wmma4(fa1, fb0, acc[1][0]);
    acc[1][1] = wmma4(fa1, fb1, acc[1][1]);
  }

  for (int mt = 0; mt < 2; ++mt)
    for (int nt = 0; nt < 2; ++nt) {
      const int n = n_base + nt * 16 + mr;
      const float bi = bias[n];
#pragma unroll
      for (int r = 0; r < 8; ++r) {
        const int m = m_base + mt * 16 + kk * 8 + r;
        const float v = acc[mt][nt][r] + bi;
        float o;
        if (MODE == 0) o = silu_f(v);
        else           o = v * gate[(size_t)m * Dsz + n];
        OUT[(size_t)m * Dsz + n] = o;
      }
    }
}

// ---------------------------------------------------------------------------
// Kernel 3: per-(b,h) projection x@w_dt^T, softplus -> ln_da, inclusive scan
// over L -> S[b,l,h].  Grid = B*H blocks, 1024 threads (one per l).
// ---------------------------------------------------------------------------
__global__ __launch_bounds__(1024)
void dtscan_kernel(const float* __restrict__ X, const float* __restrict__ w_dt,
                   const float* __restrict__ b_dt, const float* __restrict__ ln_a,
                   float* __restrict__ S) {
  __shared__ float wlds[Dsz];
  __shared__ float scan[Lsz];
  const int b = blockIdx.x >> 3;
  const int h = blockIdx.x & 7;
  const int l = threadIdx.x;

  if (l < Dsz) wlds[l] = w_dt[h * Dsz + l];
  __syncthreads();

  const float* xr = X + (size_t)(b * Lsz + l) * Dsz;
  float dot = 0.f;
#pragma unroll 4
  for (int k = 0; k < Dsz; k += 4) {
    float4 xv = *(const float4*)(xr + k);
    dot += xv.x * wlds[k] + xv.y * wlds[k + 1] + xv.z * wlds[k + 2] + xv.w * wlds[k + 3];
  }
  const float v = dot + b_dt[h];
  const float sp = (v > 20.f) ? v : log1pf(expf(v));       // softplus
  const float ld = -expf(ln_a[h]) * sp;                    // ln_da <= 0

  scan[l] = ld;
  __syncthreads();
  // Hillis-Steele inclusive scan over 1024 elements
  for (int off = 1; off < Lsz; off <<= 1) {
    float t = scan[l];
    if (l >= off) t += scan[l - off];
    __syncthreads();
    scan[l] = t;
    __syncthreads();
  }
  S[(size_t)(b * Lsz + l) * Hsz + h] = scan[l];
}

// ---------------------------------------------------------------------------
// Kernel 4: per-(b,h): C = sum_{m>h} z[b,m,h,:] + hidden[b,h,:],
// and capture zhh = z[b,h,h,:].  Grid = B*H blocks, 256 threads.
// ---------------------------------------------------------------------------
__global__ __launch_bounds__(256)
void zsum_kernel(const float* __restrict__ Z, const float* __restrict__ hidden,
                 float* __restrict__ C, float* __restrict__ zhh) {
  __shared__ float red[256];
  const int b = blockIdx.x >> 3;
  const int h = blockIdx.x & 7;
  const int i = threadIdx.x & 63;
  const int part = threadIdx.x >> 6;   // 0..3

  float s = 0.f;
  for (int m = h + 1 + part; m < Lsz; m += 4)
    s += Z[((size_t)(b * Lsz + m) * Hsz + h) * Isz + i];
  red[threadIdx.x] = s;
  __syncthreads();
  if (part == 0) {
    const float tot = red[i] + red[i + 64] + red[i + 128] + red[i + 192];
    const int ch = (b * Hsz + h) * Isz + i;
    C[ch] = tot + hidden[ch];
    zhh[ch] = Z[((size_t)(b * Lsz + h) * Hsz + h) * Isz + i];
  }
}

// ---------------------------------------------------------------------------
// Kernel 5: h = zhh + exp(S)*C, then GroupNorm over I=64 per head.
// Grid = B*L blocks, 512 threads (one per (h,i)).
// ---------------------------------------------------------------------------
__global__ __launch_bounds__(512)
void hnorm_kernel(const float* __restrict__ S, const float* __restrict__ C,
                  const float* __restrict__ zhh, const float* __restrict__ gn_w,
                  const float* __restrict__ gn_b, float* __restrict__ Hout,
                  float* __restrict__ Hnorm) {
  __shared__ float lds[Dsz];
  const int bl = blockIdx.x;         // b*L + l
  const int b = bl >> 10;
  const int t = threadIdx.x;
  const int h = t >> 6;
  const int i = t & 63;
  const int ch = (b * Hsz + h) * Isz + i;

  const float e = expf(S[(size_t)bl * Hsz + h]);
  const float hv = zhh[ch] + e * C[ch];
  Hout[(size_t)bl * Dsz + t] = hv;

  lds[t] = hv;
  __syncthreads();
  for (int s = 32; s >= 1; s >>= 1) {
    if (i < s) lds[t] += lds[t + s];
    __syncthreads();
  }
  const float mu = lds[h * 64] * (1.0f / 64.0f);
  __syncthreads();
  const float d = hv - mu;
  lds[t] = d * d;
  __syncthreads();
  for (int s = 32; s >= 1; s >>= 1) {
    if (i < s) lds[t] += lds[t + s];
    __syncthreads();
  }
  const float var = lds[h * 64] * (1.0f / 64.0f);
  const float hn = d / sqrtf(var + GN_EPS) * gn_w[h] + gn_b[h];
  Hnorm[(size_t)bl * Dsz + t] = hn;
}

// ---------------------------------------------------------------------------
// Launch
// ---------------------------------------------------------------------------
extern "C" void kernel_launch(void* const* d_in, const int* in_sizes, int n_in,
                              void* d_out, int out_size, void* d_ws, size_t ws_size,
                              hipStream_t stream) {
  const float* x      = (const float*)d_in[0];
  const float* hidden = (const float*)d_in[1];
  const float* w_z    = (const float*)d_in[2];
  const float* b_z    = (const float*)d_in[3];
  const float* w_za   = (const float*)d_in[4];
  const float* b_za   = (const float*)d_in[5];
  const float* w_y    = (const float*)d_in[6];
  const float* b_y    = (const float*)d_in[7];
  const float* w_ya   = (const float*)d_in[8];
  const float* b_ya   = (const float*)d_in[9];
  const float* w_dt   = (const float*)d_in[10];
  const float* b_dt   = (const float*)d_in[11];
  const float* ln_a   = (const float*)d_in[12];
  const float* gn_w   = (const float*)d_in[13];
  const float* gn_b   = (const float*)d_in[14];

  float* y_out = (float*)d_out;                               // [B,L,D]
  float* h_out = y_out + (size_t)Msz * Dsz;                   // [B,L,H,I]

  // Workspace layout (floats)
  float* ws   = (float*)d_ws;
  float* Z    = ws;                                           // [M,D]   8 MB
  float* GYA  = Z    + (size_t)Msz * Dsz;                     // [M,D]   8 MB
  float* HN   = GYA  + (size_t)Msz * Dsz;                     // [M,D]   8 MB
  float* Sb   = HN   + (size_t)Msz * Dsz;                     // [M,H]   128 KB
  float* Cb   = Sb   + (size_t)Msz * Hsz;                     // [B,H,I] 8 KB
  float* ZHH  = Cb   + (size_t)Bsz * Hsz * Isz;               // [B,H,I] 8 KB

  const dim3 gGemm(Msz / 64, Dsz / 64, 1);   // (64, 8)
  const dim3 bGemm(128, 1, 1);

  // 1) z = (x Wz^T + bz) * silu(x Wza^T + bza)
  gemm_gate_z_kernel<<<gGemm, bGemm, 0, stream>>>(x, w_z, b_z, w_za, b_za, Z);
  // 2) gya = silu(x Wya^T + bya)
  gemm_single_kernel<0><<<gGemm, bGemm, 0, stream>>>(x, w_ya, b_ya, nullptr, GYA);
  // 3) S = cumsum_l(-exp(ln_a) * softplus(x w_dt^T + b_dt))
  dtscan_kernel<<<dim3(Bsz * Hsz), dim3(Lsz), 0, stream>>>(x, w_dt, b_dt, ln_a, Sb);
  // 4) C = sum_{m>h} z + hidden ;  zhh = z[b,h,h,:]
  zsum_kernel<<<dim3(Bsz * Hsz), dim3(256), 0, stream>>>(Z, hidden, Cb, ZHH);
  // 5) h = zhh + exp(S)*C ; GroupNorm -> h_norm
  hnorm_kernel<<<dim3(Msz), dim3(Dsz), 0, stream>>>(Sb, Cb, ZHH, gn_w, gn_b,
                                                    h_out, HN);
  // 6) y = (h_norm Wy^T + by) * gya
  gemm_single_kernel<1><<<gGemm, bGemm, 0, stream>>>(HN, w_y, b_y, GYA, y_out);
}